// CrossAttention_34497177321848
// MI455X (gfx1250) — compile-verified
//
#include <hip/hip_runtime.h>
#include <math.h>

typedef float v2f __attribute__((ext_vector_type(2)));
typedef float v8f __attribute__((ext_vector_type(8)));

#define BATCH 4
#define CHAN  256
#define PROJD 64
#define NHEAD 4
#define HD    16
#define NPOS  4096          // 64*64
#define QKSCALE 0.25f       // HD^-0.5
#define KBLK  64            // keys processed per online-softmax step (4 tiles)
static const unsigned long long QKVSZ = 4ull * PROJD * NPOS;   // 1,048,576 floats per buffer

__device__ __forceinline__ v8f wmma_f32(v2f a, v2f b, v8f c) {
    // D = A(16x4) * B(4x16) + C(16x16), fp32
    return __builtin_amdgcn_wmma_f32_16x16x4_f32(false, a, false, b, (short)0, c, false, false);
}

struct ProjArgs {
    const float* feat[6];
    const float* w[6];
    const float* bias[6];
    float* dst;
};

// -------------------- Kernel 1: QKV projections (6 GEMMs) --------------------
// out[pj][b][p][n] = sum_c W[p][c]*feat[b][c][n] + bias[p]
__global__ void __launch_bounds__(128) ca_proj_kernel(ProjArgs args) {
    const int lane = threadIdx.x & 31;
    const int wave = threadIdx.x >> 5;       // 0..3 -> p-tile
    const int half = lane >> 4, lm = lane & 15;
    const int n0 = blockIdx.x * 16;
    const int b  = blockIdx.y;
    const int pj = blockIdx.z;
    const int p0 = wave * 16;

    const float* feat = args.feat[pj] + (size_t)b * CHAN * NPOS;
    const float* W    = args.w[pj];
    const float* bias = args.bias[pj];
    float* dst = args.dst + (size_t)pj * QKVSZ + (size_t)b * PROJD * NPOS;

    const float* wrow = W + (p0 + lm) * CHAN;    // A: row m = lm
    v8f acc = {};
    #pragma unroll 4
    for (int kc = 0; kc < CHAN; kc += 4) {
        const int ka = kc + 2 * half;
        v2f a = *(const v2f*)(wrow + ka);                       // A[m][k], A[m][k+1]
        v2f bm;
        bm.x = feat[(size_t)(ka + 0) * NPOS + n0 + lm];         // B[k][n]
        bm.y = feat[(size_t)(ka + 1) * NPOS + n0 + lm];
        acc = wmma_f32(a, bm, acc);
    }
    float bl[8];
    #pragma unroll
    for (int r = 0; r < 8; ++r) bl[r] = bias[p0 + r + 8 * half];
    #pragma unroll
    for (int r = 0; r < 8; ++r) {
        const int m = r + 8 * half;
        dst[(size_t)(p0 + m) * NPOS + n0 + lm] = acc[r] + bl[r];
    }
}

// -------------------- Kernel 2: flash cross-attention --------------------
// One wave per (branch, b, head, 16-query tile). Online softmax, 64 keys per step.
__global__ void __launch_bounds__(128) ca_attn_kernel(const float* qkv, float* attn_out) {
    __shared__ float lds_p[4][4][16 * 17];   // [wave][key-tile][query*17 + key]
    __shared__ float lds_c[4][16];
    __shared__ float lds_l[4][16];

    const int lane = threadIdx.x & 31;
    const int wave = threadIdx.x >> 5;
    const int half = lane >> 4, lm = lane & 15;
    const int br = blockIdx.z;
    const int b  = blockIdx.y >> 2;
    const int h  = blockIdx.y & 3;
    const int q0 = (blockIdx.x * 4 + wave) * 16;

    const size_t bhoff = ((size_t)b * PROJD + h * HD) * NPOS;
    const float* Q = qkv + (size_t)(br ? 3 : 0) * QKVSZ + bhoff;
    const float* K = qkv + (size_t)(br ? 4 : 1) * QKVSZ + bhoff;
    const float* V = qkv + (size_t)(br ? 5 : 2) * QKVSZ + bhoff;
    float* O = attn_out + (size_t)br * QKVSZ + bhoff;

    // A operand (Q tile), loop invariant: A[m=query][k=d]
    v2f aq[4];
    #pragma unroll
    for (int c = 0; c < 4; ++c) {
        const int d = 4 * c + 2 * half;
        aq[c].x = Q[(size_t)(d + 0) * NPOS + q0 + lm];
        aq[c].y = Q[(size_t)(d + 1) * NPOS + q0 + lm];
    }

    v8f o = {};
    float mrun[8], lrun[8];
    #pragma unroll
    for (int r = 0; r < 8; ++r) { mrun[r] = -1e30f; lrun[r] = 0.0f; }

    for (int j0 = 0; j0 < NPOS; j0 += KBLK) {
        if (j0 + KBLK < NPOS) {
            __builtin_prefetch(&K[(size_t)(2 * half) * NPOS + j0 + KBLK + lm], 0, 1);
            __builtin_prefetch(&V[(size_t)lm * NPOS + j0 + KBLK + 2 * half], 0, 1);
        }
        // 4 S tiles = Q^T K  (M=query, N=key, K=d in 4 chunks each)
        v8f s[4];
        #pragma unroll
        for (int t = 0; t < 4; ++t) {
            s[t] = {};
            #pragma unroll
            for (int c = 0; c < 4; ++c) {
                const int d = 4 * c + 2 * half;
                v2f bk;
                bk.x = K[(size_t)(d + 0) * NPOS + j0 + 16 * t + lm];
                bk.y = K[(size_t)(d + 1) * NPOS + j0 + 16 * t + lm];
                s[t] = wmma_f32(aq[c], bk, s[t]);
            }
        }
        // block-wide row max (per-lane over 4 tiles first, then one shuffle tree)
        float mnew[8], corr[8], rs[8];
        #pragma unroll
        for (int r = 0; r < 8; ++r) {
            float rm = fmaxf(fmaxf(s[0][r], s[1][r]), fmaxf(s[2][r], s[3][r])) * QKSCALE;
            rm = fmaxf(rm, __shfl_xor(rm, 1));
            rm = fmaxf(rm, __shfl_xor(rm, 2));
            rm = fmaxf(rm, __shfl_xor(rm, 4));
            rm = fmaxf(rm, __shfl_xor(rm, 8));
            mnew[r] = fmaxf(mrun[r], rm);
            corr[r] = __expf(mrun[r] - mnew[r]);
            rs[r] = 0.0f;
        }
        __syncthreads();   // previous step's LDS reads complete
        #pragma unroll
        for (int t = 0; t < 4; ++t) {
            #pragma unroll
            for (int r = 0; r < 8; ++r) {
                const float p = __expf(s[t][r] * QKSCALE - mnew[r]);
                rs[r] += p;
                lds_p[wave][t][(half * 8 + r) * 17 + lm] = p;   // [query][key]
            }
        }
        #pragma unroll
        for (int r = 0; r < 8; ++r) {
            float v = rs[r];
            v += __shfl_xor(v, 1);
            v += __shfl_xor(v, 2);
            v += __shfl_xor(v, 4);
            v += __shfl_xor(v, 8);
            lrun[r] = lrun[r] * corr[r] + v;
            mrun[r] = mnew[r];
        }
        if (lm == 0) {
            #pragma unroll
            for (int r = 0; r < 8; ++r) lds_c[wave][half * 8 + r] = corr[r];
        }
        __syncthreads();

        // rescale O (N index = query = lm) and accumulate O += V * P
        const float cq = lds_c[wave][lm];
        #pragma unroll
        for (int r = 0; r < 8; ++r) o[r] *= cq;
        #pragma unroll
        for (int t = 0; t < 4; ++t) {
            #pragma unroll
            for (int c = 0; c < 4; ++c) {
                const int jj = 4 * c + 2 * half;
                v2f av = *(const v2f*)(V + (size_t)lm * NPOS + j0 + 16 * t + jj); // A[m=d][k=key]
                v2f bp;
                bp.x = lds_p[wave][t][lm * 17 + jj + 0];                          // B[k=key][n=query]
                bp.y = lds_p[wave][t][lm * 17 + jj + 1];
                o = wmma_f32(av, bp, o);
            }
        }
    }
    __syncthreads();
    if (lm == 0) {
        #pragma unroll
        for (int r = 0; r < 8; ++r) lds_l[wave][half * 8 + r] = lrun[r];
    }
    __syncthreads();
    const float linv = 1.0f / lds_l[wave][lm];
    #pragma unroll
    for (int r = 0; r < 8; ++r)
        O[(size_t)(r + 8 * half) * NPOS + q0 + lm] = o[r] * linv;
}

// -------------------- Kernel 3: out projection + bias + residual --------------------
__global__ void __launch_bounds__(512) ca_outproj_kernel(
    const float* attn_out, const float* feat_a, const float* feat_b,
    const float* ow_a, const float* ob_a, const float* ow_b, const float* ob_b,
    float* out) {
    const int lane = threadIdx.x & 31;
    const int wave = threadIdx.x >> 5;       // 0..15 -> c-tile (= group)
    const int half = lane >> 4, lm = lane & 15;
    const int n0 = blockIdx.x * 16;
    const int b  = blockIdx.y;
    const int br = blockIdx.z;
    const int c0 = wave * 16;

    const float* X = attn_out + (size_t)br * QKVSZ + (size_t)b * PROJD * NPOS;
    const float* W = br ? ow_b : ow_a;
    const float* bias = br ? ob_b : ob_a;
    const float* F = (br ? feat_b : feat_a) + (size_t)b * CHAN * NPOS;
    float* Y = out + (size_t)br * ((size_t)BATCH * CHAN * NPOS) + (size_t)b * CHAN * NPOS;

    const float* wrow = W + (c0 + lm) * PROJD;
    v8f acc = {};
    #pragma unroll
    for (int kc = 0; kc < PROJD; kc += 4) {
        const int ka = kc + 2 * half;
        v2f a = *(const v2f*)(wrow + ka);
        v2f bx;
        bx.x = X[(size_t)(ka + 0) * NPOS + n0 + lm];
        bx.y = X[(size_t)(ka + 1) * NPOS + n0 + lm];
        acc = wmma_f32(a, bx, acc);
    }
    float bl[8], fl[8];
    #pragma unroll
    for (int r = 0; r < 8; ++r) {
        const int cc = c0 + r + 8 * half;
        bl[r] = bias[cc];
        fl[r] = F[(size_t)cc * NPOS + n0 + lm];
    }
    #pragma unroll
    for (int r = 0; r < 8; ++r) {
        const int cc = c0 + r + 8 * half;
        Y[(size_t)cc * NPOS + n0 + lm] = acc[r] + bl[r] + fl[r];
    }
}

// -------------------- Kernel 4: deterministic group stats --------------------
// one block per (branch,b,group): reduce 16 channels * 4096 positions (contiguous 64K floats)
__global__ void __launch_bounds__(256) ca_stats_kernel(const float* y, float* stats) {
    __shared__ float sh1[256];
    __shared__ float sh2[256];
    const int gidx = blockIdx.x;             // 0..127
    const int br = gidx >> 6;
    const int b  = (gidx >> 4) & 3;
    const int g  = gidx & 15;
    const float* base = y + (size_t)br * ((size_t)BATCH * CHAN * NPOS)
                          + ((size_t)b * CHAN + g * 16) * NPOS;
    float s1 = 0.0f, s2 = 0.0f;
    const int nvec = (16 * NPOS) / 4;        // 16384 float4
    for (int i = threadIdx.x; i < nvec; i += 256) {
        float4 v = ((const float4*)base)[i];
        s1 += v.x + v.y + v.z + v.w;
        s2 += v.x * v.x + v.y * v.y + v.z * v.z + v.w * v.w;
    }
    sh1[threadIdx.x] = s1; sh2[threadIdx.x] = s2;
    __syncthreads();
    for (int off = 128; off > 0; off >>= 1) {
        if (threadIdx.x < off) {
            sh1[threadIdx.x] += sh1[threadIdx.x + off];
            sh2[threadIdx.x] += sh2[threadIdx.x + off];
        }
        __syncthreads();
    }
    if (threadIdx.x == 0) {
        stats[gidx * 2 + 0] = sh1[0];
        stats[gidx * 2 + 1] = sh2[0];
    }
}

// -------------------- Kernel 5: group norm finalize (in place) --------------------
__global__ void __launch_bounds__(256) ca_norm_kernel(
    float* out, const float* stats,
    const float* g_a, const float* b_a, const float* g_b, const float* b_b) {
    const unsigned PER = (unsigned)BATCH * CHAN * NPOS;
    unsigned i = blockIdx.x * 256u + threadIdx.x;
    if (i >= 2u * PER) return;
    const unsigned br = i / PER;
    const unsigned r  = i - br * PER;
    const unsigned b  = r / (CHAN * NPOS);
    const unsigned r2 = r - b * (CHAN * NPOS);
    const unsigned c  = r2 / NPOS;
    const unsigned g  = c >> 4;
    const float* st = stats + ((br * 4 + b) * 16 + g) * 2;
    const float invn = 1.0f / 65536.0f;      // 16 channels * 4096 positions
    const float mean = st[0] * invn;
    const float var  = st[1] * invn - mean * mean;
    const float inv  = rsqrtf(var + 1e-5f);
    const float gam  = (br ? g_b : g_a)[c];
    const float bet  = (br ? b_b : b_a)[c];
    out[i] = (out[i] - mean) * inv * gam + bet;
}

extern "C" void kernel_launch(void* const* d_in, const int* in_sizes, int n_in,
                              void* d_out, int out_size, void* d_ws, size_t ws_size,
                              hipStream_t stream) {
    const float* feat_a = (const float*)d_in[0];
    const float* feat_b = (const float*)d_in[1];
    const float* qaw = (const float*)d_in[2];  const float* qab = (const float*)d_in[3];
    const float* kbw = (const float*)d_in[4];  const float* kbb = (const float*)d_in[5];
    const float* vbw = (const float*)d_in[6];  const float* vbb = (const float*)d_in[7];
    const float* qbw = (const float*)d_in[8];  const float* qbb = (const float*)d_in[9];
    const float* kaw = (const float*)d_in[10]; const float* kab = (const float*)d_in[11];
    const float* vaw = (const float*)d_in[12]; const float* vab = (const float*)d_in[13];
    const float* oaw = (const float*)d_in[14]; const float* oab = (const float*)d_in[15];
    const float* obw = (const float*)d_in[16]; const float* obb = (const float*)d_in[17];
    const float* nag = (const float*)d_in[18]; const float* nabb = (const float*)d_in[19];
    const float* nbg = (const float*)d_in[20]; const float* nbb = (const float*)d_in[21];

    float* ws    = (float*)d_ws;
    float* qkv   = ws;                       // 6 * QKVSZ floats
    float* attn  = ws + 6 * QKVSZ;           // 2 * QKVSZ floats
    float* stats = ws + 8 * QKVSZ;           // 256 floats
    float* out   = (float*)d_out;

    ProjArgs pa;
    const float* feats[6] = {feat_a, feat_b, feat_b, feat_b, feat_a, feat_a};
    const float* wsrc[6]  = {qaw, kbw, vbw, qbw, kaw, vaw};
    const float* bsrc[6]  = {qab, kbb, vbb, qbb, kab, vab};
    for (int i = 0; i < 6; ++i) { pa.feat[i] = feats[i]; pa.w[i] = wsrc[i]; pa.bias[i] = bsrc[i]; }
    pa.dst = qkv;

    dim3 gp(NPOS / 16, BATCH, 6);
    ca_proj_kernel<<<gp, 128, 0, stream>>>(pa);

    dim3 ga(NPOS / 16 / 4, BATCH * NHEAD, 2);
    ca_attn_kernel<<<ga, 128, 0, stream>>>(qkv, attn);

    dim3 go(NPOS / 16, BATCH, 2);
    ca_outproj_kernel<<<go, 512, 0, stream>>>(attn, feat_a, feat_b, oaw, oab, obw, obb, out);

    ca_stats_kernel<<<128, 256, 0, stream>>>(out, stats);

    const unsigned total = 2u * BATCH * CHAN * NPOS;
    ca_norm_kernel<<<(total + 255) / 256, 256, 0, stream>>>(out, stats, nag, nabb, nbg, nbb);
}